// SegmentationLoss_64338610094311
// MI455X (gfx1250) — compile-verified
//
#include <hip/hip_runtime.h>
#include <math.h>

namespace {

constexpr int HH    = 512;
constexpr int WWD   = 512;
constexpr int CC    = 3;
constexpr int ROWF  = WWD * CC;      // 1536 floats per image row (all channels)
constexpr int TILE  = 16;            // output rows per block
constexpr int RB    = TILE + 16;     // buffered rows incl. 8-row halo = 32
constexpr int NWRD  = WWD / 32;      // 16 packed u32 words per row per channel
constexpr int NTHR  = 384;           // 12 wave32 waves
constexpr int NTILES = HH / TILE;    // 32
constexpr int BB    = 8;

// border-flip XOR pattern: pixel flipped iff (border row) XOR (border col)
__device__ __forceinline__ unsigned xmask_row(int ry, int wi) {
  unsigned ec = (wi == 0 ? 1u : 0u) | (wi == NWRD - 1 ? 0x80000000u : 0u);
  return (ry == 0 || ry == HH - 1) ? (0xFFFFFFFFu ^ ec) : ec;
}

// W8 = per-start-position window-8 AND of the 48-bit funnel word e.
// Centered window of width 2h+1 at out bit j lives at W_w[j+8-h].
__device__ __forceinline__ unsigned extE(unsigned long long W8, int h) {
  unsigned long long w;
  if (h == 8) { unsigned long long w16 = W8 & (W8 >> 8); w = w16 & (w16 >> 1); return (unsigned)w; }
  if (h == 7) { w = W8 & (W8 >> 7); return (unsigned)(w >> 1); }
  if (h == 6) { w = W8 & (W8 >> 5); return (unsigned)(w >> 2); }
  if (h == 5) { w = W8 & (W8 >> 3); return (unsigned)(w >> 3); }
  w = W8 & (W8 >> 1); return (unsigned)(w >> 4);           // h == 4
}
__device__ __forceinline__ unsigned extD(unsigned long long W8, int h) {
  unsigned long long w;
  if (h == 8) { unsigned long long w16 = W8 | (W8 >> 8); w = w16 | (w16 >> 1); return (unsigned)w; }
  if (h == 7) { w = W8 | (W8 >> 7); return (unsigned)(w >> 1); }
  if (h == 6) { w = W8 | (W8 >> 5); return (unsigned)(w >> 2); }
  if (h == 5) { w = W8 | (W8 >> 3); return (unsigned)(w >> 3); }
  w = W8 | (W8 >> 1); return (unsigned)(w >> 4);           // h == 4
}

} // namespace

extern "C" __global__ void __launch_bounds__(NTHR)
seg_loss_main(const float* __restrict__ in_det,
              const float* __restrict__ p_det,
              const float* __restrict__ edge_w,
              float* __restrict__ partials)
{
  __shared__ unsigned           Mw [CC][RB][NWRD];      // flipped bit-mask
  __shared__ unsigned long long W8e[CC][RB][NWRD];      // window-8 AND chain
  __shared__ unsigned long long W8d[CC][RB][NWRD];      // window-8 OR  chain
  __shared__ unsigned           EDGEw[CC][TILE][NWRD];  // edge bits
  __shared__ float              rowbuf[2][ROWF];        // async staging (reused as red)

  const int t    = threadIdx.x;
  const int tile = blockIdx.x;
  const int b    = blockIdx.y;
  const int y0   = tile * TILE;
  const float* inb = in_det + (size_t)b * HH * ROWF;

  // ---------------- Phase 1: async-stage rows -> LDS, ballot bit-pack ------
  auto issue_row = [&](int r) {
    int ry = y0 - 8 + r;
    ry = ry < 0 ? 0 : (ry > HH - 1 ? HH - 1 : ry);      // clamp (data unused if OOB)
    unsigned long long ga =
        (unsigned long long)(size_t)(inb + (size_t)ry * ROWF) + (unsigned long long)(t * 16);
    unsigned la = (unsigned)(size_t)(&rowbuf[r & 1][0]) + (unsigned)(t * 16);
    // 384 lanes x 16B = one full 6144B row, direct to LDS (no VGPR round trip)
    asm volatile("global_load_async_to_lds_b128 %0, %1, off"
                 :: "v"(la), "v"(ga) : "memory");
  };

  issue_row(0);
  for (int r = 0; r < RB; ++r) {
    if (r + 1 < RB) {
      issue_row(r + 1);
      asm volatile("s_wait_asynccnt 0x1" ::: "memory");  // oldest (row r) complete
    } else {
      asm volatile("s_wait_asynccnt 0x0" ::: "memory");
    }
    __syncthreads();
    const int ry = y0 - 8 + r;
    if (ry >= 0 && ry < HH) {
      const float* rb = rowbuf[r & 1];
      {
        int x = t;                                       // waves 0..11 -> x 0..383
        float v0 = rb[3 * x + 0], v1 = rb[3 * x + 1], v2 = rb[3 * x + 2];
        unsigned w0 = __builtin_amdgcn_ballot_w32(v0 == 1.0f);
        unsigned w1 = __builtin_amdgcn_ballot_w32(v1 == 1.0f);
        unsigned w2 = __builtin_amdgcn_ballot_w32(v2 == 1.0f);
        if ((t & 31) == 0) {
          int wi = x >> 5;
          unsigned xm = xmask_row(ry, wi);
          Mw[0][r][wi] = w0 ^ xm; Mw[1][r][wi] = w1 ^ xm; Mw[2][r][wi] = w2 ^ xm;
        }
      }
      if (t < WWD - NTHR) {                              // waves 0..3 -> x 384..511
        int x = NTHR + t;
        float v0 = rb[3 * x + 0], v1 = rb[3 * x + 1], v2 = rb[3 * x + 2];
        unsigned w0 = __builtin_amdgcn_ballot_w32(v0 == 1.0f);
        unsigned w1 = __builtin_amdgcn_ballot_w32(v1 == 1.0f);
        unsigned w2 = __builtin_amdgcn_ballot_w32(v2 == 1.0f);
        if ((t & 31) == 0) {
          int wi = x >> 5;
          unsigned xm = xmask_row(ry, wi);
          Mw[0][r][wi] = w0 ^ xm; Mw[1][r][wi] = w1 ^ xm; Mw[2][r][wi] = w2 ^ xm;
        }
      }
    }
    __syncthreads();
  }

  // ------------- Phase 2: per-row horizontal window-8 AND/OR chains --------
  for (int k = 0; k < (CC * RB * NWRD) / NTHR; ++k) {    // 4 iters
    int item = t + NTHR * k;
    int c = item % 3;
    int rest = item / 3;
    int i = rest & (NWRD - 1);
    int r = rest >> 4;
    int ry = y0 - 8 + r;
    if (ry >= 0 && ry < HH) {
      unsigned mcu = Mw[c][r][i];
      unsigned mp  = (i > 0)        ? Mw[c][r][i - 1] : 0u;
      unsigned mn  = (i < NWRD - 1) ? Mw[c][r][i + 1] : 0u;
      unsigned mpE = (i > 0)        ? mp : 0xFFFFFFFFu;  // x-OOB neutral
      unsigned mnE = (i < NWRD - 1) ? mn : 0xFFFFFFFFu;
      unsigned long long eE = (unsigned long long)(mpE >> 24)
                            | ((unsigned long long)mcu << 8)
                            | ((unsigned long long)(mnE & 0xFFu) << 40);
      unsigned long long eD = (unsigned long long)(mp >> 24)
                            | ((unsigned long long)mcu << 8)
                            | ((unsigned long long)(mn & 0xFFu) << 40);
      unsigned long long a2 = eE & (eE >> 1);
      unsigned long long a4 = a2 & (a2 >> 2);
      W8e[c][r][i] = a4 & (a4 >> 4);
      unsigned long long o2 = eD | (eD >> 1);
      unsigned long long o4 = o2 | (o2 >> 2);
      W8d[c][r][i] = o4 | (o4 >> 4);
    }
  }
  __syncthreads();

  // ------------- Phase 3: vertical combine with ellipse row widths ---------
  const int htab[9] = {8, 8, 8, 7, 7, 6, 5, 4, 0};       // halfwidth per |dy|
  for (int k = 0; k < (CC * TILE * NWRD) / NTHR; ++k) {  // 2 iters
    int item = t + NTHR * k;
    int c = item % 3;
    int rest = item / 3;
    int i = rest & (NWRD - 1);
    int oy = rest >> 4;
    unsigned E = 0xFFFFFFFFu, D = 0u;
#pragma unroll
    for (int d = -8; d <= 8; ++d) {
      int ry = y0 + oy + d;
      if (ry < 0 || ry >= HH) continue;                  // y-OOB neutral
      int rr = oy + 8 + d;
      int h = htab[d < 0 ? -d : d];
      if (h == 0) {
        unsigned m = Mw[c][rr][i];
        E &= m; D |= m;
      } else {
        E &= extE(W8e[c][rr][i], h);
        D |= extD(W8d[c][rr][i], h);
      }
    }
    EDGEw[c][oy][i] = D & ~E;                            // morphological gradient
  }
  __syncthreads();

  // ------------- Phase 4: stream p_detections, weighted accumulation -------
  const int   c   = t % 3;                               // fixed: 384 % 3 == 0
  const float ewc = edge_w[c];
  float aw = 0.f, awp = 0.f;
  const float* pb = p_det + (size_t)b * HH * ROWF;
  for (int oy = 0; oy < TILE; ++oy) {
    const int ry = y0 + oy;
    const float* prow = pb + (size_t)ry * ROWF;
    const int borderY = (ry == 0) | (ry == HH - 1);
#pragma unroll
    for (int k = 0; k < ROWF / NTHR; ++k) {              // 4 coalesced loads/row
      int f = t + NTHR * k;
      int x = f / 3;
      float p = prow[f];
      unsigned mbit = (Mw[c][8 + oy][x >> 5] >> (x & 31)) & 1u;
      int flip = borderY ^ (int)((x == 0) | (x == WWD - 1));
      unsigned obit = mbit ^ (unsigned)flip;             // recover original mask
      float base = obit ? (1.0f - p) : p;
      float rr = fmaxf(0.0f, base);
      float pix = rr * rr;
      unsigned eb = (EDGEw[c][oy][x >> 5] >> (x & 31)) & 1u;
      float w = eb ? ewc : 1.0f;                         // 1+(ew-1)*edge, exact
      aw  += w;
      awp += w * pix;
    }
  }

  // ------------- deterministic per-block reduction (rowbuf reused) ---------
  __syncthreads();
  float* red = &rowbuf[0][0];
  red[t] = aw;
  red[NTHR + t] = awp;
  __syncthreads();
  if (t < CC) {
    float sw = 0.f, swp = 0.f;
    for (int j = t; j < NTHR; j += CC) { sw += red[j]; swp += red[NTHR + j]; }
    int idx = ((b * CC + t) * NTILES + tile) * 2;
    partials[idx + 0] = sw;
    partials[idx + 1] = swp;
  }
}

extern "C" __global__ void __launch_bounds__(32)
seg_loss_finalize(const float* __restrict__ partials,
                  const float* __restrict__ in_masks,
                  const float* __restrict__ logvars,
                  float* __restrict__ out)
{
  __shared__ float vals[BB * CC];
  int t = threadIdx.x;
  if (t < BB * CC) {
    int b = t / 3, c = t % 3;
    const float* p = partials + (size_t)(b * 3 + c) * NTILES * 2;
    float sw = 0.f, swp = 0.f;
    for (int i = 0; i < NTILES; ++i) { sw += p[2 * i]; swp += p[2 * i + 1]; }
    float raw = swp / sw;                                // sum(w*pix)/sum(w)
    float lv = logvars[c];
    vals[t] = in_masks[b * 3 + c] * (expf(-lv) * raw + lv);
  }
  __syncthreads();
  if (t < BB)
    out[t] = (vals[3 * t] + vals[3 * t + 1] + vals[3 * t + 2]) * (1.0f / 3.0f);
}

extern "C" void kernel_launch(void* const* d_in, const int* in_sizes, int n_in,
                              void* d_out, int out_size, void* d_ws, size_t ws_size,
                              hipStream_t stream) {
  (void)in_sizes; (void)n_in; (void)out_size; (void)ws_size;
  const float* in_masks = (const float*)d_in[0];
  const float* in_det   = (const float*)d_in[1];
  const float* p_det    = (const float*)d_in[2];
  const float* edge_w   = (const float*)d_in[3];
  const float* logvars  = (const float*)d_in[4];
  float* out      = (float*)d_out;
  float* partials = (float*)d_ws;  // 8*3*32*2 floats = 6 KB, fully overwritten

  dim3 grid(NTILES, BB);           // 256 blocks x 12 waves
  seg_loss_main<<<grid, NTHR, 0, stream>>>(in_det, p_det, edge_w, partials);
  seg_loss_finalize<<<1, 32, 0, stream>>>(partials, in_masks, logvars, out);
}